// Sent_DAN_75849122448044
// MI455X (gfx1250) — compile-verified
//
#include <hip/hip_runtime.h>
#include <hip/hip_bf16.h>
#include <math.h>

#define BB   64
#define SS   2048
#define DD   1024
#define HH   300
#define SSEG 16
#define SEGLEN (SS / SSEG)

typedef __attribute__((ext_vector_type(2))) float v2f;
typedef __attribute__((ext_vector_type(8))) float v8f;

// ---------------------------------------------------------------------------
// Kernel 1: end_ind[i] = first s with ids[i][s]==1, else fallback BB (=shape[0])
// LDS atomicMin is order-independent -> deterministic.
// ---------------------------------------------------------------------------
__global__ void find_end_kernel(const int* __restrict__ ids,
                                int* __restrict__ end_ind) {
    __shared__ int mn;
    const int i = blockIdx.x;
    if (threadIdx.x == 0) mn = 0x7fffffff;
    __syncthreads();
    int local = 0x7fffffff;
    for (int s = threadIdx.x; s < SS; s += blockDim.x) {
        if (ids[i * SS + s] == 1 && s < local) local = s;
    }
    atomicMin(&mn, local);
    __syncthreads();
    if (threadIdx.x == 0) end_ind[i] = (mn == 0x7fffffff) ? BB : mn;
}

// ---------------------------------------------------------------------------
// Kernel 2: partial[seg][i][d] = sum_{s in seg, s<end} inputs[i][s][d]*drop[i][s]
// 256 threads * float4 = one full D row per iteration (4 KB coalesced b128).
// Loop bound clipped at end_ind -> no wasted HBM traffic past EOS.
// Every partial slot is written (zero if segment fully past end) -> no init.
// ---------------------------------------------------------------------------
__global__ void partial_sum_kernel(const float* __restrict__ inputs,
                                   const float* __restrict__ drop,
                                   const int* __restrict__ end_ind,
                                   float* __restrict__ partial) {
    const int i   = blockIdx.x;     // batch row
    const int seg = blockIdx.y;     // S segment
    const int t   = threadIdx.x;    // float4 column, 0..255
    const int end = end_ind[i];
    const int s0  = seg * SEGLEN;
    const int s1  = min(s0 + SEGLEN, end);

    float4 acc = make_float4(0.f, 0.f, 0.f, 0.f);
    const float4* in4 = (const float4*)(inputs + (size_t)i * SS * DD);
    const float*  dm  = drop + (size_t)i * SS;
    for (int s = s0; s < s1; ++s) {
        const float  w = dm[s];
        const float4 v = in4[(size_t)s * (DD / 4) + t];
        acc.x = fmaf(w, v.x, acc.x);
        acc.y = fmaf(w, v.y, acc.y);
        acc.z = fmaf(w, v.z, acc.z);
        acc.w = fmaf(w, v.w, acc.w);
    }
    float4* p4 = (float4*)(partial + ((size_t)seg * BB + i) * DD);
    p4[t] = acc;
}

// ---------------------------------------------------------------------------
// Kernel 3: euph[i][d] = (1/end_ind[i]) * sum_seg partial[seg][i][d]
// Fixed summation order -> deterministic.
// ---------------------------------------------------------------------------
__global__ void final_sum_kernel(const float* __restrict__ partial,
                                 const int* __restrict__ end_ind,
                                 float* __restrict__ euph) {
    const int gid = blockIdx.x * blockDim.x + threadIdx.x;  // 0..16383
    const int i = gid / (DD / 4);
    const int t = gid % (DD / 4);
    float4 acc = make_float4(0.f, 0.f, 0.f, 0.f);
    for (int seg = 0; seg < SSEG; ++seg) {
        const float4 v =
            ((const float4*)(partial + ((size_t)seg * BB + i) * DD))[t];
        acc.x += v.x; acc.y += v.y; acc.z += v.z; acc.w += v.w;
    }
    const float inv = 1.0f / (float)end_ind[i];
    acc.x *= inv; acc.y *= inv; acc.z *= inv; acc.w *= inv;
    ((float4*)(euph + (size_t)i * DD))[t] = acc;
}

// ---------------------------------------------------------------------------
// Kernel 4: hidden = tanh(euph @ W1^T + b1) via full-fp32 WMMA 16x16x4.
// One wave per 16x16 output tile, chaining 256 K=4 WMMAs.
// A (16x4): lanes 0-15 -> K={0,1}, lanes 16-31 -> K={2,3}, M=lane&15.
// B (4x16): same K split, N=lane&15 (column of W1^T = row of W1).
// N >= 300 handled by clamp + zero-mask (EXEC stays all-1s through WMMA).
// ---------------------------------------------------------------------------
__global__ void mlp1_wmma_kernel(const float* __restrict__ euph,
                                 const float* __restrict__ W1,
                                 const float* __restrict__ b1,
                                 float* __restrict__ hidden) {
    const int mtile = blockIdx.x;           // 0..3  (M = 64)
    const int ntile = blockIdx.y;           // 0..18 (N = 304 padded)
    const int lane  = threadIdx.x;          // 0..31
    const int half  = lane >> 4;
    const int l     = lane & 15;

    const int   m     = mtile * 16 + l;
    const int   n     = ntile * 16 + l;
    const int   nc    = (n < HH) ? n : (HH - 1);
    const float nmask = (n < HH) ? 1.f : 0.f;

    const float* arow = euph + (size_t)m * DD;
    const float* brow = W1   + (size_t)nc * DD;

    v8f acc = {};
    for (int kk = 0; kk < DD / 4; ++kk) {
        const int k = kk * 4 + half * 2;
        v2f a; a.x = arow[k];          a.y = arow[k + 1];
        v2f b; b.x = brow[k] * nmask;  b.y = brow[k + 1] * nmask;
        // (neg_a, A, neg_b, B, c_mod, C, reuse_a, reuse_b)
        acc = __builtin_amdgcn_wmma_f32_16x16x4_f32(
            false, a, false, b, (short)0, acc, false, false);
    }

    if (n < HH) {
        const float bias = b1[n];
        #pragma unroll
        for (int r = 0; r < 8; ++r) {
            const int mo = mtile * 16 + half * 8 + r;   // C layout: VGPR r
            hidden[(size_t)mo * HH + n] = tanhf(acc[r] + bias);
        }
    }
}

// ---------------------------------------------------------------------------
// Kernel 5: out[i][j] = hidden[i] . W2[j] + b2[j]   (64x2, trivially small)
// ---------------------------------------------------------------------------
__global__ void mlp2_kernel(const float* __restrict__ hidden,
                            const float* __restrict__ W2,
                            const float* __restrict__ b2,
                            float* __restrict__ out) {
    const int t = threadIdx.x;   // 0..127
    const int i = t >> 1;
    const int j = t & 1;
    float s = b2[j];
    const float* hrow = hidden + (size_t)i * HH;
    const float* wrow = W2     + (size_t)j * HH;
    for (int h = 0; h < HH; ++h) s = fmaf(hrow[h], wrow[h], s);
    out[i * 2 + j] = s;
}

// ---------------------------------------------------------------------------
extern "C" void kernel_launch(void* const* d_in, const int* in_sizes, int n_in,
                              void* d_out, int out_size, void* d_ws, size_t ws_size,
                              hipStream_t stream) {
    const float* inputs = (const float*)d_in[0];   // [64,2048,1024] f32
    const int*   ids    = (const int*)  d_in[1];   // [64,2048] int
    const float* drop   = (const float*)d_in[2];   // [64,2048,1] f32
    const float* W1     = (const float*)d_in[3];   // [300,1024]
    const float* b1     = (const float*)d_in[4];   // [300]
    const float* W2     = (const float*)d_in[5];   // [2,300]
    const float* b2     = (const float*)d_in[6];   // [2]
    float*       out    = (float*)d_out;           // [64,2]

    char* ws = (char*)d_ws;
    const size_t partial_bytes = (size_t)SSEG * BB * DD * sizeof(float); // 4 MB
    const size_t euph_bytes    = (size_t)BB * DD * sizeof(float);        // 256 KB
    const size_t hidden_bytes  = (size_t)BB * HH * sizeof(float);        // 75 KB
    float* partial = (float*)(ws);
    float* euph    = (float*)(ws + partial_bytes);
    float* hidden  = (float*)(ws + partial_bytes + euph_bytes);
    int*   end_ind = (int*)  (ws + partial_bytes + euph_bytes + hidden_bytes);

    find_end_kernel<<<BB, 256, 0, stream>>>(ids, end_ind);
    partial_sum_kernel<<<dim3(BB, SSEG), 256, 0, stream>>>(inputs, drop, end_ind, partial);
    final_sum_kernel<<<(BB * DD / 4) / 256, 256, 0, stream>>>(partial, end_ind, euph);
    mlp1_wmma_kernel<<<dim3(BB / 16, (HH + 15) / 16), 32, 0, stream>>>(euph, W1, b1, hidden);
    mlp2_kernel<<<1, 128, 0, stream>>>(hidden, W2, b2, out);
}